// RelationalLoRAGINConv_81209241633070
// MI455X (gfx1250) — compile-verified
//
#include <hip/hip_runtime.h>
#include <hip/hip_bf16.h>

typedef __attribute__((ext_vector_type(2))) float v2f;
typedef __attribute__((ext_vector_type(4))) float v4f;
typedef __attribute__((ext_vector_type(8))) float v8f;

#define NN 100000
#define DD 128
#define RR 8
#define NRELS 16
#define EE 250000

// ---------------------------------------------------------------- zero kernel
__global__ void k_zero(float* __restrict__ p, int n) {
    int i = blockIdx.x * blockDim.x + threadIdx.x;
    if (i < n) p[i] = 0.0f;
}

// ---------------------------------------------------------------- edge kernel
// Per wave: 16-edge tile. base = x_j @ W^T via v_wmma_f32_16x16x4_f32,
// adapter = A_r (B_r^T x_j) via VALU (rank-8), fused scatter-atomics.
#define EK_WAVES 4

__global__ void __launch_bounds__(EK_WAVES * 32)
k_edge(const float* __restrict__ x, const int* __restrict__ ei,
       const int* __restrict__ et, const float* __restrict__ W,
       const float* __restrict__ A_emb, const float* __restrict__ B_emb,
       float* __restrict__ aggr) {
    __shared__ float sX[EK_WAVES][16 * 132];   // gathered x_j rows, padded
    __shared__ float sBj[EK_WAVES][16 * 8];    // per-edge rank-8 projection
    __shared__ int sSrc[EK_WAVES][16];
    __shared__ int sDst[EK_WAVES][16];
    __shared__ int sR[EK_WAVES][16];
    __shared__ int sVal[EK_WAVES][16];

    const int w = threadIdx.x >> 5;
    const int lane = threadIdx.x & 31;
    const int laneN = lane & 15;
    const int hi = lane >> 4;

    const int tile = blockIdx.x * EK_WAVES + w;
    const int e0 = tile * 16;

    // --- tile metadata (clamped at tail) ---
    if (lane < 16) {
        int e = e0 + lane;
        int valid = (e < EE) ? 1 : 0;
        int ec = valid ? e : (EE - 1);
        sSrc[w][lane] = ei[ec];
        sDst[w][lane] = ei[EE + ec];
        sR[w][lane]   = et[ec];
        sVal[w][lane] = valid;
    }
    __syncthreads();

    // --- gather 16 rows of x into LDS (L2-resident gather) ---
#pragma unroll 4
    for (int m = 0; m < 16; ++m) {
        int srcRow = sSrc[w][m];
        v4f xv = *((const v4f*)(x + (size_t)srcRow * DD) + lane);
        *((v4f*)(&sX[w][m * 132]) + lane) = xv;
    }
    __syncthreads();

    // --- Bj[m][k] = sum_d B_r[d][k] * x_j[m][d]  (lane -> (m, 4 k's)) ---
    {
        const int m = laneN;
        const int kg = hi * 4;
        const int r = sR[w][m];
        const float* Bbase = B_emb + r * (DD * RR) + kg;
        const float* xrow = &sX[w][m * 132];
        v4f acc4 = {};
#pragma unroll 8
        for (int d = 0; d < DD; ++d) {
            float xv = xrow[d];
            v4f bv = *(const v4f*)(Bbase + d * RR);
            acc4 += xv * bv;
        }
        *(v4f*)(&sBj[w][m * RR + kg]) = acc4;
    }
    __syncthreads();

    // --- base: [16 x 128] @ W^T, 8 N-tiles, K=128 in 32 wmma steps ---
    v8f acc[8];
    {
        v8f z = {};
#pragma unroll
        for (int nt = 0; nt < 8; ++nt) acc[nt] = z;
    }
    const float* xw = &sX[w][0];
    for (int kk = 0; kk < DD / 4; ++kk) {
        int kb = kk * 4 + hi * 2;
        v2f a;
        a[0] = xw[laneN * 132 + kb];
        a[1] = xw[laneN * 132 + kb + 1];
#pragma unroll
        for (int nt = 0; nt < 8; ++nt) {
            int n = nt * 16 + laneN;
            const float* wp = W + n * DD + kb;   // B(k,n) = W[n,k]
            v2f b;
            b[0] = wp[0];
            b[1] = wp[1];
            acc[nt] = __builtin_amdgcn_wmma_f32_16x16x4_f32(
                false, a, false, b, (short)0, acc[nt], false, false);
        }
    }

    // --- adapter fold + fused scatter: one atomic per (edge, channel) ---
#pragma unroll
    for (int nt = 0; nt < 8; ++nt) {
        int n = nt * 16 + laneN;
#pragma unroll
        for (int v = 0; v < 8; ++v) {
            int m = v + 8 * hi;               // D-frag row for this lane/vgpr
            int r = sR[w][m];
            const float* Ar = A_emb + r * (DD * RR) + n * RR;
            v4f a0 = *(const v4f*)(Ar);
            v4f a1 = *(const v4f*)(Ar + 4);
            const float* bj = &sBj[w][m * RR];
            v4f j0 = *(const v4f*)(bj);
            v4f j1 = *(const v4f*)(bj + 4);
            v4f p = a0 * j0 + a1 * j1;
            float adp = p[0] + p[1] + p[2] + p[3];
            float val = acc[nt][v] + adp;
            if (sVal[w][m]) {
                unsafeAtomicAdd(aggr + (size_t)sDst[w][m] * DD + n, val);
            }
        }
    }
}

// ---------------------------------------------------------------- MLP kernel
#define MK_WAVES 2
#define SROW 260   // padded LDS row stride (floats)

template <int KDIM, int NT, bool RELU>
__device__ __forceinline__ void mlp_layer(const float* __restrict__ Wl,
                                          const float* __restrict__ bl,
                                          float* sbuf, int lane) {
    const int laneN = lane & 15;
    const int hi = lane >> 4;
    v8f acc[NT];
    {
        v8f z = {};
#pragma unroll
        for (int nt = 0; nt < NT; ++nt) acc[nt] = z;
    }
    for (int kk = 0; kk < KDIM / 4; ++kk) {
        int kb = kk * 4 + hi * 2;
        v2f a;
        a[0] = sbuf[laneN * SROW + kb];
        a[1] = sbuf[laneN * SROW + kb + 1];
#pragma unroll
        for (int nt = 0; nt < NT; ++nt) {
            int n = nt * 16 + laneN;
            const float* wp = Wl + n * KDIM + kb;   // B(k,n) = Wl[n,k]
            v2f b;
            b[0] = wp[0];
            b[1] = wp[1];
            acc[nt] = __builtin_amdgcn_wmma_f32_16x16x4_f32(
                false, a, false, b, (short)0, acc[nt], false, false);
        }
    }
    __syncthreads();   // all A-frag reads of this layer complete block-wide
#pragma unroll
    for (int nt = 0; nt < NT; ++nt) {
        int n = nt * 16 + laneN;
        float bb = bl[n];
#pragma unroll
        for (int v = 0; v < 8; ++v) {
            float val = acc[nt][v] + bb;
            if (RELU) val = fmaxf(val, 0.0f);
            sbuf[(v + 8 * hi) * SROW + n] = val;
        }
    }
    __syncthreads();   // writes visible before next layer reads
}

__global__ void __launch_bounds__(MK_WAVES * 32)
k_mlp(const float* __restrict__ x, const float* __restrict__ aggr,
      const float* __restrict__ epsp,
      const float* __restrict__ W1, const float* __restrict__ b1,
      const float* __restrict__ W2, const float* __restrict__ b2,
      const float* __restrict__ W3, const float* __restrict__ b3,
      float* __restrict__ out) {
    __shared__ float sBuf[MK_WAVES][16 * SROW];

    const int w = threadIdx.x >> 5;
    const int lane = threadIdx.x & 31;
    const int tile = blockIdx.x * MK_WAVES + w;
    const int node0 = tile * 16;
    const float s = 1.0f + epsp[0];
    float* sbuf = &sBuf[w][0];

    // --- GIN update: (1+eps)*x + aggr -> LDS activation buffer ---
#pragma unroll 4
    for (int m = 0; m < 16; ++m) {
        int row = node0 + m;
        if (row > NN - 1) row = NN - 1;
        v4f xv = *((const v4f*)(x + (size_t)row * DD) + lane);
        v4f av = *((const v4f*)(aggr + (size_t)row * DD) + lane);
        v4f o = xv * s + av;
        *((v4f*)(sbuf + m * SROW) + lane) = o;
    }
    __syncthreads();

    mlp_layer<128, 16, true >(W1, b1, sbuf, lane);   // 128 -> 256, relu
    mlp_layer<256, 16, true >(W2, b2, sbuf, lane);   // 256 -> 256, relu
    mlp_layer<256,  8, false>(W3, b3, sbuf, lane);   // 256 -> 128

    // --- coalesced masked store ---
#pragma unroll 4
    for (int m = 0; m < 16; ++m) {
        int row = node0 + m;
        if (row < NN) {
            v4f o = *((const v4f*)(sbuf + m * SROW) + lane);
            *((v4f*)(out + (size_t)row * DD) + lane) = o;
        }
    }
}

// ---------------------------------------------------------------- launcher
extern "C" void kernel_launch(void* const* d_in, const int* in_sizes, int n_in,
                              void* d_out, int out_size, void* d_ws, size_t ws_size,
                              hipStream_t stream) {
    const float* x      = (const float*)d_in[0];
    const int*   ei     = (const int*)d_in[1];
    const int*   et     = (const int*)d_in[2];
    const float* W      = (const float*)d_in[3];
    const float* eps    = (const float*)d_in[4];
    const float* A_emb  = (const float*)d_in[5];
    const float* B_emb  = (const float*)d_in[6];
    const float* W1     = (const float*)d_in[7];
    const float* b1     = (const float*)d_in[8];
    const float* W2     = (const float*)d_in[9];
    const float* b2     = (const float*)d_in[10];
    const float* W3     = (const float*)d_in[11];
    const float* b3     = (const float*)d_in[12];
    float* out  = (float*)d_out;
    float* aggr = (float*)d_ws;   // [N, D] scatter accumulator

    // 1) zero the accumulator (harness does not re-zero between replays)
    {
        int n = NN * DD;
        int blocks = (n + 255) / 256;
        k_zero<<<blocks, 256, 0, stream>>>(aggr, n);
    }
    // 2) fused edge messages (WMMA base + rank-8 adapter) + scatter-atomics
    {
        int edgesPerBlock = EK_WAVES * 16;
        int blocks = (EE + edgesPerBlock - 1) / edgesPerBlock;
        k_edge<<<blocks, EK_WAVES * 32, 0, stream>>>(x, ei, et, W, A_emb, B_emb, aggr);
    }
    // 3) fused GIN update + 3-layer WMMA MLP
    {
        int nodesPerBlock = MK_WAVES * 16;
        int blocks = (NN + nodesPerBlock - 1) / nodesPerBlock;
        k_mlp<<<blocks, MK_WAVES * 32, 0, stream>>>(x, aggr, eps, W1, b1, W2, b2, W3, b3, out);
    }
}